// MainNet_16329465659865
// MI455X (gfx1250) — compile-verified
//
#include <hip/hip_runtime.h>
#include <math.h>

// MI455X / gfx1250, wave32. Memory-bound hypernetwork forward:
// per (b,j): h0 = relu(x·W0+b0); h1 = relu(h0·W1+b1) [256x256, dominant];
// out = h1·W2+b2; then means / sqrt(exp+eps) / sparsemax.
// Layer-1 matvec runs through V_WMMA_F32_16X16X4_F32 (A row M=0 = h0 chunk,
// B tile = streamed W1 4x16 block). Branch-free A build (v_cndmask, no exec
// churn); each wave processes two adjacent 16-col tiles per k-step sharing A.

typedef __attribute__((ext_vector_type(2))) float v2f;
typedef __attribute__((ext_vector_type(8))) float v8f;

#define B_   128
#define H_   256
#define G_   8
#define OD_  24
#define J_   7
#define EPS_ 1e-7f
#define TP_  512680   // TOTAL_PARAMS per sample

__global__ __launch_bounds__(256) void mainnet_kernel(
    const float* __restrict__ x,       // (B, 8)
    const float* __restrict__ w,       // (B, TP_)
    float* __restrict__ out)           // (3, J, B, G)
{
    __shared__ float h0s[H_];
    __shared__ float h1s[H_];
    __shared__ float p2[8 * 32];       // layer-2 partials (8 waves x 24 used)
    __shared__ float outv[OD_];

    const int tid  = threadIdx.x;
    const int lane = tid & 31;
    const int wave = tid >> 5;
    const int j    = blockIdx.x % J_;
    const int b    = blockIdx.x / J_;
    const int in_dim = j + 1;

    // layer offset within a sample: sum_{m<j} (256*(m+1) + 72216)
    const size_t layer_off = (size_t)(128 * j * (j + 1) + 72216 * j);
    const float* wp = w + (size_t)b * TP_ + layer_off;
    const float* W0 = wp;
    const float* b0 = W0 + (size_t)in_dim * H_;
    const float* W1 = b0 + H_;
    const float* b1 = W1 + (size_t)H_ * H_;
    const float* W2 = b1 + H_;
    const float* b2 = W2 + (size_t)H_ * OD_;

    // ---------- Layer 0: h0[tid] = relu(b0 + sum_i x[i]*W0[i,tid]) ----------
    {
        float acc = b0[tid];
        #pragma unroll
        for (int i = 0; i < 8; ++i) {
            if (i < in_dim) acc = fmaf(x[b * 8 + i], W0[(size_t)i * H_ + tid], acc);
        }
        h0s[tid] = fmaxf(acc, 0.0f);
    }
    // Prefetch layer-2 weights (24KB) while layer 1 streams (global_prefetch_b8).
    __builtin_prefetch(W2 + (size_t)tid * OD_, 0, 0);
    __syncthreads();

    // ---------- Layer 1: h1 = relu(h0 · W1 + b1) via V_WMMA_F32_16X16X4_F32 ----
    // A (16x4): only row M=0 valid (= h0[k0..k0+3]); documented layout:
    //   VGPR0 lanes0-15:K=0, lanes16-31:K=2 ; VGPR1: K=1, K=3.
    //   -> row M=0 lives in lanes 0 (K=0,1) and 16 (K=2,3).
    // B (4x16): mirrored layout:
    //   VGPR0 lanes0-15:row k0, lanes16-31:row k0+2 ; VGPR1: rows k0+1, k0+3.
    // Each wave owns 32 output columns (two N-tiles) sharing one A per k-step.
    {
        const int  lhi = lane >> 4;          // lane half: 0 or 1
        const int  lo  = lane & 15;          // column within tile / M row id
        const bool m0  = (lo == 0);          // lanes carrying A row M=0
        const int  o0  = wave * 32;

        v8f c0 = {};
        v8f c1 = {};
        #pragma unroll 4
        for (int k0 = 0; k0 < H_; k0 += 4) {
            const int kb = k0 + (lhi << 1);            // even -> 8B aligned
            // Broadcast LDS read of h0 chunk (2 addresses per wave, no conflict)
            const v2f ha = *(const v2f*)&h0s[kb];
            v2f am;
            am.x = m0 ? ha.x : 0.0f;                   // v_cndmask, branch-free
            am.y = m0 ? ha.y : 0.0f;
            // B tiles: rows kb, kb+1; cols [o0, o0+32) -> contiguous 128B/row
            const float* r = W1 + (size_t)kb * H_ + o0 + lo;
            v2f bm0, bm1;
            bm0.x = r[0];        bm0.y = r[H_];
            bm1.x = r[16];       bm1.y = r[H_ + 16];
            c0 = __builtin_amdgcn_wmma_f32_16x16x4_f32(
                     false, am, false, bm0, (short)0, c0, false, false);
            c1 = __builtin_amdgcn_wmma_f32_16x16x4_f32(
                     false, am, false, bm1, (short)0, c1, false, false);
        }
        // D row M=0 -> element 0 on lanes 0..15 (N = lane)
        if (lane < 16) {
            float v0 = c0[0] + b1[o0 + lane];
            float v1 = c1[0] + b1[o0 + 16 + lane];
            h1s[o0 + lane]      = fmaxf(v0, 0.0f);
            h1s[o0 + 16 + lane] = fmaxf(v1, 0.0f);
        }
    }
    __syncthreads();

    // ---------- Layer 2: out[o] = b2[o] + sum_i h1[i]*W2[i,o], o < 24 ----------
    {
        if (lane < OD_) {
            float acc = 0.0f;
            const int i0 = wave * 32;
            #pragma unroll 8
            for (int i = i0; i < i0 + 32; ++i)
                acc = fmaf(h1s[i], W2[(size_t)i * OD_ + lane], acc);
            p2[wave * 32 + lane] = acc;
        }
    }
    __syncthreads();
    if (tid < OD_) {
        float acc = b2[tid];
        #pragma unroll
        for (int wv = 0; wv < 8; ++wv) acc += p2[wv * 32 + tid];
        outv[tid] = acc;
    }
    __syncthreads();

    // ---------- Heads: means, scales, sparsemax ----------
    const size_t obase = (size_t)j * (B_ * G_) + (size_t)b * G_;
    const size_t plane = (size_t)(J_ * B_ * G_);
    if (tid < G_) {
        out[0 * plane + obase + tid] = outv[tid];
        float v = __expf(outv[G_ + tid]) + EPS_;
        out[1 * plane + obase + tid] = sqrtf(v);
    }
    if (tid == 0) {
        float z[G_], zs[G_];
        #pragma unroll
        for (int g = 0; g < G_; ++g) { z[g] = outv[2 * G_ + g]; zs[g] = z[g]; }
        // register-only bubble sort, descending (fully unrolled, constant indices)
        #pragma unroll
        for (int a2 = 0; a2 < G_; ++a2) {
            #pragma unroll
            for (int b2 = 0; b2 < G_ - 1; ++b2) {
                if (zs[b2] < zs[b2 + 1]) { float tf = zs[b2]; zs[b2] = zs[b2 + 1]; zs[b2 + 1] = tf; }
            }
        }
        // k_z = #{k : 1 + k*zs[k-1] > cs[k-1]}, tau = (cs[k_z-1]-1)/k_z
        float csarr[G_];
        {
            float cs = 0.0f;
            #pragma unroll
            for (int kk = 0; kk < G_; ++kk) { cs += zs[kk]; csarr[kk] = cs; }
        }
        int kz = 0;
        #pragma unroll
        for (int kk = 1; kk <= G_; ++kk)
            if (1.0f + (float)kk * zs[kk - 1] > csarr[kk - 1]) kz++;
        float cs_k = 0.0f;
        #pragma unroll
        for (int kk = 0; kk < G_; ++kk)
            if (kk == kz - 1) cs_k = csarr[kk];
        const float tau = (cs_k - 1.0f) / (float)kz;
        #pragma unroll
        for (int g = 0; g < G_; ++g)
            out[2 * plane + obase + g] = fmaxf(z[g] - tau, 0.0f);
    }
}

extern "C" void kernel_launch(void* const* d_in, const int* in_sizes, int n_in,
                              void* d_out, int out_size, void* d_ws, size_t ws_size,
                              hipStream_t stream) {
    const float* x = (const float*)d_in[0];   // (128, 8) f32
    const float* w = (const float*)d_in[1];   // (128, 512680) f32
    float* out = (float*)d_out;               // 3*7*128*8 f32
    dim3 grid(B_ * J_);
    dim3 block(256);
    hipLaunchKernelGGL(mainnet_kernel, grid, block, 0, stream, x, w, out);
}